// CosineSimilarity_9388798509363
// MI455X (gfx1250) — compile-verified
//
#include <hip/hip_runtime.h>
#include <hip/hip_bf16.h>
#include <math.h>

// ---------------------------------------------------------------------------
// Cosine similarity, MI455X (gfx1250, wave32).
//
// Roofline: 256 MB streamed once -> ~11.5 us at 23.3 TB/s; ~0.4 GFLOP of math.
// Purely bandwidth bound. Strategy: b128 non-temporal loads (single-pass
// stream, don't pollute the 192 MB L2), 8192 waves in flight, and the three
// running reductions (a.b, a.a, b.b) folded through V_WMMA_F32_16X16X4_F32
// with an all-ones B matrix:  D[m][n] = sum_k A[m,k]  -- i.e. the WMMA unit
// performs the cross-lane summation for us, accumulating in the C operand.
// Element placement inside A is irrelevant for a total sum, so each lane just
// contributes its own products.
// ---------------------------------------------------------------------------

typedef __attribute__((ext_vector_type(2))) float v2f;
typedef __attribute__((ext_vector_type(4))) float v4f;
typedef __attribute__((ext_vector_type(8))) float v8f;

#define TPB 512                 // 16 waves per workgroup
#define HALFWAVES (TPB / 16)    // lanes {0,16} of each wave spill -> 32 partials

__global__ __launch_bounds__(TPB) void cosine_stage1(
    const float* __restrict__ x1,
    const float* __restrict__ x2,
    float* __restrict__ cos_ws,
    int D)
{
    const int s = blockIdx.x;
    const size_t base = (size_t)s * (size_t)D;
    const float* __restrict__ a = x1 + base;
    const float* __restrict__ b = x2 + base;
    const int tid = threadIdx.x;

    // Three 16x16 f32 accumulators (8 VGPRs each).
    v8f accAB = {};
    v8f accAA = {};
    v8f accBB = {};
    const v2f ones = {1.0f, 1.0f};   // B matrix: 4x16 of 1.0 across the wave

    const int stride = TPB * 4;      // elements consumed per iteration
    const int iters  = D / stride;   // 65536 / 2048 = 32 for the reference shape
    int idx = tid * 4;

    #pragma unroll 4
    for (int i = 0; i < iters; ++i, idx += stride) {
        // Coalesced 16B-aligned b128 loads, non-temporal (stream-once data).
        v4f av = __builtin_nontemporal_load((const v4f*)(a + idx));
        v4f bv = __builtin_nontemporal_load((const v4f*)(b + idx));

        // Fold each float4 pair down to 2 partial products per reduction.
        v2f pab = { __builtin_fmaf(av.z, bv.z, av.x * bv.x),
                    __builtin_fmaf(av.w, bv.w, av.y * bv.y) };
        v2f paa = { __builtin_fmaf(av.z, av.z, av.x * av.x),
                    __builtin_fmaf(av.w, av.w, av.y * av.y) };
        v2f pbb = { __builtin_fmaf(bv.z, bv.z, bv.x * bv.x),
                    __builtin_fmaf(bv.w, bv.w, bv.y * bv.y) };

        // D = A x Ones + C : the XDL unit sums the 64 per-wave products.
        // 8 args: (neg_a, A, neg_b, B, c_mod, C, reuse_a, reuse_b)
        accAB = __builtin_amdgcn_wmma_f32_16x16x4_f32(
            false, pab, false, ones, (short)0, accAB, false, false);
        accAA = __builtin_amdgcn_wmma_f32_16x16x4_f32(
            false, paa, false, ones, (short)0, accAA, false, false);
        accBB = __builtin_amdgcn_wmma_f32_16x16x4_f32(
            false, pbb, false, ones, (short)0, accBB, false, false);
    }

    // Each lane holds column N = lane%16 of D; all columns are identical
    // (B was all ones). Summing the 8 VGPRs gives half a column (M=0..7 on
    // lanes 0-15, M=8..15 on lanes 16-31), so lane 0 + lane 16 of each wave
    // together hold the full wave partial. Spill those two to LDS; no
    // shuffles needed, and all WMMA work is done before any divergence.
    float sAB = accAB[0] + accAB[1] + accAB[2] + accAB[3]
              + accAB[4] + accAB[5] + accAB[6] + accAB[7];
    float sAA = accAA[0] + accAA[1] + accAA[2] + accAA[3]
              + accAA[4] + accAA[5] + accAA[6] + accAA[7];
    float sBB = accBB[0] + accBB[1] + accBB[2] + accBB[3]
              + accBB[4] + accBB[5] + accBB[6] + accBB[7];

    __shared__ float red[3][HALFWAVES];
    if ((tid & 15) == 0) {               // lanes 0 and 16 of every wave
        const int slot = tid >> 4;       // 0 .. HALFWAVES-1
        red[0][slot] = sAB;
        red[1][slot] = sAA;
        red[2][slot] = sBB;
    }
    __syncthreads();

    if (tid == 0) {
        float dot = 0.f, naa = 0.f, nbb = 0.f;
        #pragma unroll
        for (int w = 0; w < HALFWAVES; ++w) {
            dot += red[0][w];
            naa += red[1][w];
            nbb += red[2][w];
        }
        const float denom = sqrtf(naa) * sqrtf(nbb);
        cos_ws[s] = dot / fmaxf(denom, 1e-8f);   // eps clamp per reference
    }
}

// Segmented mean over consecutive groups of n. n lives in device memory
// (graph capture forbids host reads), so index with it on-device.
__global__ void cosine_stage2(const float* __restrict__ cos_ws,
                              const int* __restrict__ n_ptr,
                              float* __restrict__ out,
                              int out_n)
{
    const int t = blockIdx.x * blockDim.x + threadIdx.x;
    if (t >= out_n) return;
    const int n = n_ptr[0];
    float s = 0.f;
    for (int i = 0; i < n; ++i) s += cos_ws[t * n + i];
    out[t] = s / (float)n;
}

extern "C" void kernel_launch(void* const* d_in, const int* in_sizes, int n_in,
                              void* d_out, int out_size, void* d_ws, size_t ws_size,
                              hipStream_t stream)
{
    const float* x1    = (const float*)d_in[0];
    const float* x2    = (const float*)d_in[1];
    const int*   n_ptr = (const int*)d_in[2];
    float* cos_ws = (float*)d_ws;          // 512 floats of scratch
    float* out    = (float*)d_out;

    const int D = 256 * 256;               // per-sample flattened length (L*C)
    const int N = in_sizes[0] / D;         // 512 samples

    cosine_stage1<<<N, TPB, 0, stream>>>(x1, x2, cos_ws, D);

    const int t2 = 64;
    cosine_stage2<<<(out_size + t2 - 1) / t2, t2, 0, stream>>>(
        cos_ws, n_ptr, out, out_size);
}